// Yolov1_Loss_58669253263945
// MI455X (gfx1250) — compile-verified
//
#include <hip/hip_runtime.h>
#include <cstdint>

#define CH 30
#define CELLS_PER_BLOCK 256
#define BLOCK 256
#define NWAVES (BLOCK / 32)
#define IMGF 448.0f
#define GRIDF 64.0f
#define EPSF 1e-8f
#define L_COORD 5.0f
#define L_NOOBJ 0.5f

__device__ __forceinline__ void wave_reduce6(float v[6]) {
#pragma unroll
  for (int off = 16; off > 0; off >>= 1) {
#pragma unroll
    for (int j = 0; j < 6; ++j) v[j] += __shfl_down(v[j], off, 32);
  }
}

__global__ __launch_bounds__(BLOCK) void yolo_partial_kernel(
    const float* __restrict__ bb, const float* __restrict__ gt,
    float* __restrict__ partials, int ncells) {
  __shared__ __align__(16) float lds_bb[CELLS_PER_BLOCK * CH];
  __shared__ __align__(16) float lds_gt[CELLS_PER_BLOCK * CH];
  __shared__ float wsum[NWAVES][6];

  const int tid = threadIdx.x;
  const int cell0 = blockIdx.x * CELLS_PER_BLOCK;

  // ---- Async global -> LDS staging (CDNA5 async-tensor path) ----
  // 256 cells * 30 f32 = 30720 B per buffer = 1920 b128 transfers.
  const uint64_t bb_base = (uint64_t)(uintptr_t)(bb + (size_t)cell0 * CH);
  const uint64_t gt_base = (uint64_t)(uintptr_t)(gt + (size_t)cell0 * CH);
  const uint32_t lds_bb_base = (uint32_t)(uintptr_t)(&lds_bb[0]);
  const uint32_t lds_gt_base = (uint32_t)(uintptr_t)(&lds_gt[0]);

  const int n128 = CELLS_PER_BLOCK * CH * 4 / 16;  // 1920
  for (int i = tid; i < n128; i += BLOCK) {
    uint32_t off = (uint32_t)i * 16u;
    uint32_t dst_bb = lds_bb_base + off;
    uint32_t dst_gt = lds_gt_base + off;
    asm volatile("global_load_async_to_lds_b128 %0, %1, %2"
                 :: "v"(dst_bb), "v"(off), "s"(bb_base) : "memory");
    asm volatile("global_load_async_to_lds_b128 %0, %1, %2"
                 :: "v"(dst_gt), "v"(off), "s"(gt_base) : "memory");
  }
  asm volatile("s_wait_asynccnt 0" ::: "memory");
  __syncthreads();

  // ---- Per-cell compute ----
  float v[6] = {0.f, 0.f, 0.f, 0.f, 0.f, 0.f};
  const int cell = cell0 + tid;
  if (cell < ncells) {
    float b[CH], g[CH];
#pragma unroll
    for (int c = 0; c < CH; ++c) b[c] = lds_bb[tid * CH + c];
#pragma unroll
    for (int c = 0; c < CH; ++c) g[c] = lds_gt[tid * CH + c];

    const int x = cell % 7;
    const int y = (cell / 7) % 7;

    const bool second = b[4] < b[9];
    const float other_conf = second ? b[4] : b[9];
    const float pb_conf    = second ? b[9] : b[4];
    const float px = second ? b[5] : b[0];
    const float py = second ? b[6] : b[1];
    const float pw = second ? b[7] : b[2];
    const float ph = second ? b[8] : b[3];

    const bool obj = (g[9] != 0.0f);

    const float conf_other = L_NOOBJ * other_conf * other_conf;
    const float conf_noobj = L_NOOBJ * pb_conf * pb_conf;

    // coordinate loss
    const float dx = g[0] - px;
    const float dy = g[1] - py;
    const float dw = sqrtf(g[2] + EPSF) - sqrtf(pw + EPSF);
    const float dh = sqrtf(g[3] + EPSF) - sqrtf(ph + EPSF);
    const float coord = L_COORD * (dx * dx + dy * dy + dw * dw + dh * dh);

    const float dconf = g[4] - pb_conf;
    const float conf_obj = dconf * dconf;

    float sumsq = 0.0f;
#pragma unroll
    for (int c = 0; c < 20; ++c) {
      const float d = g[10 + c] - b[10 + c];
      sumsq = fmaf(d, d, sumsq);
    }
    const float classes = (sumsq * (1.0f / 20.0f)) * 20.0f;  // mirror mean()*C

    // IoU term
    const float gx = (float)x * GRIDF;
    const float gy = (float)y * GRIDF;
    const float pcx = truncf(gx + px * GRIDF);
    const float pcy = truncf(gy + py * GRIDF);
    const float pwi = truncf(pw * IMGF);
    const float phi = truncf(ph * IMGF);
    const float plx = fmaxf(0.0f, pcx - pwi * 0.5f);
    const float puy = fmaxf(0.0f, pcy - phi * 0.5f);
    const float prx = fminf(IMGF - 1.0f, pcx + pwi * 0.5f);
    const float pdy = fminf(IMGF - 1.0f, pcy + phi * 0.5f);
    const float pA = (prx - plx) * (pdy - puy);
    const float glx = g[5], guy = g[6], grx = g[7], gdy = g[8];
    const float gA = (grx - glx) * (gdy - guy);
    const float clx = fmaxf(plx, glx);
    const float crx = fminf(prx, grx);
    const float cuy = fmaxf(puy, guy);
    const float cdy = fminf(pdy, gdy);
    const float inter = (crx - clx) * (cdy - cuy);
    const float uni = pA + gA - inter;
    const bool valid = (crx >= clx) && (cdy >= cuy);
    const float iou = valid ? (inter / fmaxf(uni, 1e-9f)) : 0.0f;

    // v: [loss, loss_coord, loss_confidence, loss_classes, iou_sum, object_num]
    if (obj) {
      v[0] = conf_other + coord + conf_obj + classes;
      v[1] = coord;
      v[2] = conf_other + conf_obj;
      v[3] = classes;
      v[4] = iou;
      v[5] = 1.0f;
    } else {
      v[0] = conf_other + conf_noobj;
      v[2] = conf_other + conf_noobj;
    }
  }

  // ---- Wave32 shuffle reduction + per-block partial ----
  wave_reduce6(v);
  const int lane = tid & 31;
  const int wave = tid >> 5;
  if (lane == 0) {
#pragma unroll
    for (int j = 0; j < 6; ++j) wsum[wave][j] = v[j];
  }
  __syncthreads();
  if (tid == 0) {
    float t[6] = {0.f, 0.f, 0.f, 0.f, 0.f, 0.f};
    for (int w = 0; w < NWAVES; ++w)
#pragma unroll
      for (int j = 0; j < 6; ++j) t[j] += wsum[w][j];
#pragma unroll
    for (int j = 0; j < 6; ++j) partials[(size_t)blockIdx.x * 6 + j] = t[j];
  }
}

__global__ __launch_bounds__(BLOCK) void yolo_final_kernel(
    const float* __restrict__ partials, int nblocks, float* __restrict__ out) {
  __shared__ float wsum[NWAVES][6];
  const int tid = threadIdx.x;
  float v[6] = {0.f, 0.f, 0.f, 0.f, 0.f, 0.f};
  for (int i = tid; i < nblocks; i += BLOCK) {
#pragma unroll
    for (int j = 0; j < 6; ++j) v[j] += partials[(size_t)i * 6 + j];
  }
  wave_reduce6(v);
  const int lane = tid & 31;
  const int wave = tid >> 5;
  if (lane == 0) {
#pragma unroll
    for (int j = 0; j < 6; ++j) wsum[wave][j] = v[j];
  }
  __syncthreads();
  if (tid == 0) {
    float t[6] = {0.f, 0.f, 0.f, 0.f, 0.f, 0.f};
    for (int w = 0; w < NWAVES; ++w)
#pragma unroll
      for (int j = 0; j < 6; ++j) t[j] += wsum[w][j];
#pragma unroll
    for (int j = 0; j < 6; ++j) out[j] = t[j];
  }
}

extern "C" void kernel_launch(void* const* d_in, const int* in_sizes, int n_in,
                              void* d_out, int out_size, void* d_ws, size_t ws_size,
                              hipStream_t stream) {
  const float* bb = (const float*)d_in[0];
  const float* gt = (const float*)d_in[1];
  float* out = (float*)d_out;
  float* partials = (float*)d_ws;

  const int ncells = in_sizes[0] / CH;                       // 802816
  const int nblocks = (ncells + CELLS_PER_BLOCK - 1) / CELLS_PER_BLOCK;  // 3136

  yolo_partial_kernel<<<nblocks, BLOCK, 0, stream>>>(bb, gt, partials, ncells);
  yolo_final_kernel<<<1, BLOCK, 0, stream>>>(partials, nblocks, out);
}